// OscillatoryChunking_19439021981974
// MI455X (gfx1250) — compile-verified
//
#include <hip/hip_runtime.h>

typedef __attribute__((ext_vector_type(2))) float v2f;
typedef __attribute__((ext_vector_type(8))) float v8f;

#define B_DIM 8
#define T_DIM 4096
#define D_DIM 1024
#define L_DIM 3
#define O_DIM 4
#define N_USED 12
#define TWO_PI_F 6.28318530717958647692f
#define DT_F 0.1f
#define NUM_STEPS_K 10
#define THRESH_F 0.3f

// ---- LDS layout for kernel 1 (all dynamic; dynamic segment base = LDS 0) ----
// [0, 12288) floats  : W K-pairs  Wp[d/2][12][2] : Wp[dp][n][p] = W[n][2*dp+p]
// [12288, ...)       : per-wave A double buffers, 16 rows x 68 floats each
#define WPAIR_FLOATS   (D_DIM / 2 * N_USED * 2)   // 12288 floats = 48 KB
#define KCHUNK         64
#define A_ROW_STRIDE   68                          // floats; 272 B, 16B aligned
#define A_BUF_FLOATS   (16 * A_ROW_STRIDE)         // 1088 floats = 4352 B
#define A_BUF_BYTES    (A_BUF_FLOATS * 4)
#define LDS_BYTES      (WPAIR_FLOATS * 4 + 8 * 2 * A_BUF_BYTES)   // 118784

// ---------------------------------------------------------------------------
// Async-copy one K chunk (16 rows x 64 floats) of the A panel into LDS.
// Each lane moves 8 x b128 blocks: row = 2*i + (lane>>4), 16B column = lane&15.
// Tracked by ASYNCcnt (8 per chunk).
// ---------------------------------------------------------------------------
__device__ __forceinline__ void async_copy_chunk(const float* gbase,
                                                 unsigned ldsByteBase,
                                                 int lane)
{
    const unsigned colb  = (unsigned)(lane & 15) * 16u;   // byte offset in row
    const int      rhalf = lane >> 4;                     // 0 or 1
    #pragma unroll
    for (int i = 0; i < 8; ++i) {
        const int row = 2 * i + rhalf;
        const unsigned long long ga =
            (unsigned long long)(const void*)gbase +
            (unsigned long long)row * (D_DIM * 4) + colb;
        const unsigned la = ldsByteBase + (unsigned)row * (A_ROW_STRIDE * 4) + colb;
        asm volatile("global_load_async_to_lds_b128 %0, %1, off"
                     :: "v"(la), "v"(ga) : "memory");
    }
}

// ---------------------------------------------------------------------------
// Kernel 1: proj[l][b][t][o] = sum_d x[b][t][d] * W[l][o][d]
// GEMM M=B*T=32768, K=1024, N=12 (cols 12..15 are discarded clamped columns).
// One wave = one 16-row M tile. A panel streams through LDS via async DMA
// (double-buffered K chunks of 64), B operand pairs come from LDS, and the
// MACs are V_WMMA_F32_16X16X4_F32.
// ---------------------------------------------------------------------------
__global__ __launch_bounds__(256) void proj_wmma_kernel(
    const float* __restrict__ x,   // [B*T, D]
    const float* __restrict__ W,   // [L*O, D]
    float* __restrict__ proj)      // [L, B, T, O]
{
    extern __shared__ float smem[];

    const int tid = threadIdx.x;

    // Stage W into LDS as K-pairs: smem[(dp*12 + n)*2 + p] = W[n*D + 2*dp + p]
    for (int i = tid; i < WPAIR_FLOATS; i += 256) {
        const int dp  = i / 24;
        const int r   = i - dp * 24;
        const int n   = r >> 1;
        const int par = r & 1;
        smem[i] = W[n * D_DIM + 2 * dp + par];
    }

    const int wave  = tid >> 5;
    const int lane  = tid & 31;
    const int mtile = blockIdx.x * 8 + wave;          // 0..2047
    const long row0 = (long)mtile * 16;               // base row in B*T
    const int  m     = lane & 15;                     // A row / B,C,D column
    const int  khalf = (lane >> 4) * 2;               // 0 or 2
    const int  nB    = (m < N_USED) ? m : (N_USED - 1);  // clamped B column

    const float*  xtile  = x + row0 * (long)D_DIM;
    const unsigned aByte0 = (unsigned)(WPAIR_FLOATS * 4) + (unsigned)wave * 2u * A_BUF_BYTES;
    const int      aFlt0  = WPAIR_FLOATS + wave * 2 * A_BUF_FLOATS;

    // Prime the pipeline: chunk 0 into buffer 0 (disjoint from W region).
    async_copy_chunk(xtile, aByte0, lane);

    __syncthreads();   // W staging visible to all waves

    v8f acc = {};
    const int NCHUNK = D_DIM / KCHUNK;                // 16
    for (int c = 0; c < NCHUNK; ++c) {
        const int buf = c & 1;
        if (c + 1 < NCHUNK) {
            // Recycled buffer's previous ds reads must have retired, then
            // overlap next chunk's DMA with this chunk's WMMAs.
            asm volatile("s_wait_dscnt 0x0" ::: "memory");
            async_copy_chunk(xtile + (c + 1) * KCHUNK,
                             aByte0 + (unsigned)((c + 1) & 1) * A_BUF_BYTES, lane);
            asm volatile("s_wait_asynccnt 0x8" ::: "memory");   // chunk c landed
        } else {
            asm volatile("s_wait_asynccnt 0x0" ::: "memory");
        }

        const int aBase = aFlt0 + buf * A_BUF_FLOATS + m * A_ROW_STRIDE + khalf;
        const int kGlob = c * KCHUNK + khalf;
        #pragma unroll
        for (int kk = 0; kk < KCHUNK; kk += 4) {
            v2f a  = *(const v2f*)&smem[aBase + kk];
            v2f bm = *(const v2f*)&smem[((kGlob + kk) >> 1) * 24 + nB * 2];
            acc = __builtin_amdgcn_wmma_f32_16x16x4_f32(
                /*neg_a=*/false, a, /*neg_b=*/false, bm,
                /*c_mod=*/(short)0, acc, /*reuse_a=*/false, /*reuse_b=*/false);
        }
    }

    // D layout: vgpr r -> row = row0 + r + (lane>>4)*8, col = lane&15
    if (m < N_USED) {
        const int l = m >> 2;
        const int o = m & 3;
        #pragma unroll
        for (int r = 0; r < 8; ++r) {
            const long row = row0 + r + (long)(lane >> 4) * 8;
            const int  b = (int)(row >> 12);          // / T_DIM
            const int  t = (int)(row & (T_DIM - 1));
            proj[(((long)(l * B_DIM + b) * T_DIM) + t) * O_DIM + o] = acc[r];
        }
    }
}

// ---------------------------------------------------------------------------
// Kernel 2: phases init (atan2 of rolled proj, mod 2pi) + 10 Kuramoto steps
// with T-axis circular-mean reduction per (l,b,o); writes phases (in place
// over proj) and level_coherences. One block per (l,b), 512 threads,
// 8 t-values x 4 oscillators held in registers per thread.
// ---------------------------------------------------------------------------
__global__ __launch_bounds__(512) void kuramoto_kernel(
    const float* __restrict__ omegas,   // [L, O]
    const float* __restrict__ Ks,       // [L]
    float* __restrict__ phases,         // [L, B, T, O] (holds proj on entry)
    float* __restrict__ lc_out)         // [B, L]
{
    const int lb = blockIdx.x;            // 0..23
    const int l  = lb / B_DIM;
    const int b  = lb - l * B_DIM;
    const int tid = threadIdx.x;
    const int TPT = T_DIM / 512;          // 8

    float* p = phases + (long)lb * T_DIM * O_DIM;

    float ph[8][4];
    #pragma unroll
    for (int i = 0; i < TPT; ++i) {
        const int t = tid * TPT + i;
        float pr[4];
        #pragma unroll
        for (int o = 0; o < 4; ++o) pr[o] = p[t * 4 + o];
        #pragma unroll
        for (int o = 0; o < 4; ++o) {
            const float prev = pr[(o + 3) & 3] + 1e-8f;
            float a = atan2f(pr[o], prev);
            a = fmodf(a, TWO_PI_F);
            if (a < 0.0f) a += TWO_PI_F;
            ph[i][o] = a;
        }
    }

    float om[4];
    #pragma unroll
    for (int o = 0; o < 4; ++o) om[o] = omegas[l * 4 + o];
    const float K = Ks[l];

    __shared__ float red[8];   // [0..3] = sum sin per o, [4..7] = sum cos per o

    for (int step = 0; step < NUM_STEPS_K; ++step) {
        float ls[4] = {0, 0, 0, 0}, lcs[4] = {0, 0, 0, 0};
        #pragma unroll
        for (int i = 0; i < TPT; ++i)
            #pragma unroll
            for (int o = 0; o < 4; ++o) {
                float s, c;
                __sincosf(ph[i][o], &s, &c);
                ls[o] += s; lcs[o] += c;
            }
        // wave32 tree reduce
        #pragma unroll
        for (int off = 16; off > 0; off >>= 1)
            #pragma unroll
            for (int o = 0; o < 4; ++o) {
                ls[o]  += __shfl_xor(ls[o],  off, 32);
                lcs[o] += __shfl_xor(lcs[o], off, 32);
            }
        if (tid < 8) red[tid] = 0.0f;
        __syncthreads();
        if ((tid & 31) == 0)
            #pragma unroll
            for (int o = 0; o < 4; ++o) {
                atomicAdd(&red[o],     ls[o]);
                atomicAdd(&red[4 + o], lcs[o]);
            }
        __syncthreads();
        float mp[4];
        #pragma unroll
        for (int o = 0; o < 4; ++o) mp[o] = atan2f(red[o], red[4 + o]);
        __syncthreads();   // protect red before next-iter reset

        #pragma unroll
        for (int i = 0; i < TPT; ++i)
            #pragma unroll
            for (int o = 0; o < 4; ++o) {
                const float coupling = K * __sinf(mp[o] - ph[i][o]) * (1.0f / (float)T_DIM);
                float np = ph[i][o] + DT_F * (om[o] + coupling);
                np = fmodf(np, TWO_PI_F);
                if (np < 0.0f) np += TWO_PI_F;
                ph[i][o] = np;
            }
    }

    // write final phases
    #pragma unroll
    for (int i = 0; i < TPT; ++i) {
        const int t = tid * TPT + i;
        #pragma unroll
        for (int o = 0; o < 4; ++o) p[t * 4 + o] = ph[i][o];
    }

    // level coherence: R_o = sqrt(mean_s^2 + mean_c^2), mean over o
    {
        float ls[4] = {0, 0, 0, 0}, lcs[4] = {0, 0, 0, 0};
        #pragma unroll
        for (int i = 0; i < TPT; ++i)
            #pragma unroll
            for (int o = 0; o < 4; ++o) {
                float s, c;
                __sincosf(ph[i][o], &s, &c);
                ls[o] += s; lcs[o] += c;
            }
        #pragma unroll
        for (int off = 16; off > 0; off >>= 1)
            #pragma unroll
            for (int o = 0; o < 4; ++o) {
                ls[o]  += __shfl_xor(ls[o],  off, 32);
                lcs[o] += __shfl_xor(lcs[o], off, 32);
            }
        if (tid < 8) red[tid] = 0.0f;
        __syncthreads();
        if ((tid & 31) == 0)
            #pragma unroll
            for (int o = 0; o < 4; ++o) {
                atomicAdd(&red[o],     ls[o]);
                atomicAdd(&red[4 + o], lcs[o]);
            }
        __syncthreads();
        if (tid == 0) {
            const float inv = 1.0f / (float)T_DIM;
            float acc = 0.0f;
            #pragma unroll
            for (int o = 0; o < 4; ++o) {
                const float ms = red[o] * inv, mc = red[4 + o] * inv;
                acc += sqrtf(ms * ms + mc * mc);
            }
            lc_out[b * L_DIM + l] = acc * 0.25f;   // (B, L) layout (transposed)
        }
    }
}

// ---------------------------------------------------------------------------
// Kernel 3: 3-tap moving coherence + local-minimum boundary detection.
// mask shape (L, B, T-2), front-padded by 2: mask[j] (j>=2) tests
// coh[j-1] < coh[j-2]-TH && coh[j-1] < coh[j]-TH, where coh[i] uses s/c over
// t = i..i+2. Recompute sin/cos locally from phases (cheap, bandwidth-light).
// ---------------------------------------------------------------------------
__global__ void boundary_kernel(const float* __restrict__ phases,
                                float* __restrict__ mask)
{
    const int TM = T_DIM - 2;                       // 4094
    const int total = L_DIM * B_DIM * TM;           // 98256
    const int idx = blockIdx.x * blockDim.x + threadIdx.x;
    if (idx >= total) return;
    const int lb = idx / TM;
    const int j  = idx - lb * TM;

    float outv = 0.0f;
    if (j >= 2) {
        const float* p = phases + (long)lb * T_DIM * O_DIM;
        float coh[3];
        #pragma unroll
        for (int q = 0; q < 3; ++q) {
            const int i = j - 2 + q;
            float acc = 0.0f;
            #pragma unroll
            for (int o = 0; o < 4; ++o) {
                float ss = 0.0f, cc = 0.0f;
                #pragma unroll
                for (int dtap = 0; dtap < 3; ++dtap) {
                    float s, c;
                    __sincosf(p[(i + dtap) * 4 + o], &s, &c);
                    ss += s; cc += c;
                }
                ss *= (1.0f / 3.0f); cc *= (1.0f / 3.0f);
                acc += sqrtf(ss * ss + cc * cc);
            }
            coh[q] = acc * 0.25f;
        }
        outv = (coh[1] < coh[0] - THRESH_F && coh[1] < coh[2] - THRESH_F)
                   ? 1.0f : 0.0f;
    }
    mask[idx] = outv;
}

// ---------------------------------------------------------------------------
extern "C" void kernel_launch(void* const* d_in, const int* in_sizes, int n_in,
                              void* d_out, int out_size, void* d_ws, size_t ws_size,
                              hipStream_t stream) {
    (void)in_sizes; (void)n_in; (void)out_size; (void)d_ws; (void)ws_size;
    const float* x      = (const float*)d_in[0];   // (B, T, D)
    const float* W      = (const float*)d_in[1];   // (L, O, D)
    const float* omegas = (const float*)d_in[2];   // (L, O)
    const float* Ks     = (const float*)d_in[3];   // (L,)

    float* out = (float*)d_out;
    float* phases = out;                                         // 393216 floats
    float* lc     = out + (long)L_DIM * B_DIM * T_DIM * O_DIM;   // 24
    float* maskp  = lc + B_DIM * L_DIM;                          // 98256

    // 1) projection GEMM -> proj stored into the phases region (in-place later)
    proj_wmma_kernel<<<256, 256, LDS_BYTES, stream>>>(x, W, phases);

    // 2) phases + Kuramoto + level coherence (in-place over proj)
    kuramoto_kernel<<<L_DIM * B_DIM, 512, 0, stream>>>(omegas, Ks, phases, lc);

    // 3) boundary mask
    const int total = L_DIM * B_DIM * (T_DIM - 2);
    boundary_kernel<<<(total + 255) / 256, 256, 0, stream>>>(phases, maskp);
}